// GCN_7172595384347
// MI455X (gfx1250) — compile-verified
//
#include <hip/hip_runtime.h>
#include <hip/hip_bf16.h>

#define NN 50000
#define NE 640000
#define CIN 128
#define HID 256
#define NG 64

typedef __attribute__((ext_vector_type(2))) float v2f;
typedef __attribute__((ext_vector_type(8))) float v8f;

// ---------------- degree / norm ----------------
__global__ void k_deg_init(float* deg) {
    int i = blockIdx.x * blockDim.x + threadIdx.x;
    if (i < NN) deg[i] = 1.0f;                 // self-loop
}
__global__ void k_deg_edges(const int* __restrict__ col, float* deg) {
    int e = blockIdx.x * blockDim.x + threadIdx.x;
    if (e < NE) atomicAdd(&deg[col[e]], 1.0f);
}
__global__ void k_dinv(float* deg) {
    int i = blockIdx.x * blockDim.x + threadIdx.x;
    if (i < NN) deg[i] = rsqrtf(deg[i]);       // deg >= 1 always
}

// ---------------- weight transpose: W[KxM] -> Wt[MxK] ----------------
__global__ void k_wt(const float* __restrict__ W, float* __restrict__ Wt, int K, int M) {
    int idx = blockIdx.x * blockDim.x + threadIdx.x;
    if (idx < K * M) {
        int k = idx / M, m = idx % M;
        Wt[(size_t)m * K + k] = W[idx];
    }
}

// ------- fp32 WMMA GEMM fused with self-loop + bias epilogue -------
// hw[N x 256] = act(A[N x K]) @ W      (act = relu if RELU)
// h [N x 256] = dinv[i]^2 * hw[i] + bias   (aggregation buffer init)
// A and h are distinct buffers (ping-pong) -> no cross-wave WAR hazard.
// Wt is W transposed [256 x K]. One wave -> 16x64 output strip.
// A-frag (16x4 f32): lane<16 row=lane K={0,1}; lane>=16 row=lane-16 K={2,3}.
// B-frag (4x16 f32): same K split, col = lane&15.
// C/D (16x16 f32, 8 VGPRs): row = (lane>>4)*8 + v, col = lane&15.
template <int K, int RELU>
__global__ void __launch_bounds__(256)
k_gemm_fused(const float* __restrict__ A, const float* __restrict__ Wt,
             const float* __restrict__ dinv, const float* __restrict__ bias,
             float* __restrict__ hw, float* __restrict__ h, int totWaves) {
    int wave = blockIdx.x * 8 + (threadIdx.x >> 5);
    if (wave >= totWaves) return;              // wave-uniform: EXEC stays all-ones
    int lane   = threadIdx.x & 31;
    int rlane  = lane & 15;
    int kOff   = (lane >> 4) * 2;              // 0 or 2
    int mBase  = (lane >> 4) * 8;              // store row base
    int rowTil = wave >> 2;
    int colGrp = (wave & 3) * 64;

    const float* arow = A + (size_t)(rowTil * 16 + rlane) * K;
    v8f acc[4] = {};

#pragma unroll 4
    for (int kt = 0; kt < K; kt += 4) {
        v2f a = *reinterpret_cast<const v2f*>(arow + kt + kOff);
        if (RELU) { a[0] = fmaxf(a[0], 0.0f); a[1] = fmaxf(a[1], 0.0f); }
#pragma unroll
        for (int j = 0; j < 4; ++j) {
            const v2f b = *reinterpret_cast<const v2f*>(
                Wt + (size_t)(colGrp + j * 16 + rlane) * K + kt + kOff);
            acc[j] = __builtin_amdgcn_wmma_f32_16x16x4_f32(
                false, a, false, b, (short)0, acc[j], false, false);
        }
    }

    // epilogue: write transformed features and self-loop-initialized aggregate
    float d2[8];
#pragma unroll
    for (int v = 0; v < 8; ++v) {
        float d = dinv[rowTil * 16 + mBase + v];
        d2[v] = d * d;
    }
#pragma unroll
    for (int j = 0; j < 4; ++j) {
        float bc = bias[colGrp + j * 16 + rlane];
#pragma unroll
        for (int v = 0; v < 8; ++v) {
            size_t idx = (size_t)(rowTil * 16 + mBase + v) * HID + colGrp + j * 16 + rlane;
            float val = acc[j][v];
            hw[idx] = val;
            h[idx]  = d2[v] * val + bc;
        }
    }
}

// ------- edge scatter: h[col] += dinv[row]*dinv[col]*hw[row] -------
// 64 lanes per edge, 4 channels per lane (b128 gather + 4 f32 atomics).
__global__ void __launch_bounds__(256)
k_edge_agg(const int* __restrict__ row, const int* __restrict__ col,
           const float* __restrict__ dinv, const float* __restrict__ hw,
           float* __restrict__ h) {
    int slot = threadIdx.x >> 6;               // 4 edge slots per block
    int c4   = (threadIdx.x & 63) * 4;         // channel base
    for (int e = blockIdx.x * 4 + slot; e < NE; e += gridDim.x * 4) {
        int r = row[e], t = col[e];
        float nrm = dinv[r] * dinv[t];
        const float4 v = *reinterpret_cast<const float4*>(&hw[(size_t)r * HID + c4]);
        float* dst = &h[(size_t)t * HID + c4];
        atomicAdd(dst + 0, nrm * v.x);
        atomicAdd(dst + 1, nrm * v.y);
        atomicAdd(dst + 2, nrm * v.z);
        atomicAdd(dst + 3, nrm * v.w);
    }
}

// ---------------- pooling ----------------
__global__ void k_pool_zero(float* sums, float* cnts) {
    int i = blockIdx.x * blockDim.x + threadIdx.x;
    if (i < NG * HID) sums[i] = 0.0f;
    if (i < NG) cnts[i] = 0.0f;
}
__global__ void __launch_bounds__(256)
k_pool_sum(const int* __restrict__ batch, const float* __restrict__ h,
           float* sums, float* cnts) {
    int slot = threadIdx.x >> 6;               // 4 node slots per block
    int c4   = (threadIdx.x & 63) * 4;
    for (int i = blockIdx.x * 4 + slot; i < NN; i += gridDim.x * 4) {
        int g = batch[i];
        const float4 v = *reinterpret_cast<const float4*>(&h[(size_t)i * HID + c4]);
        float* dst = &sums[g * HID + c4];
        atomicAdd(dst + 0, v.x);
        atomicAdd(dst + 1, v.y);
        atomicAdd(dst + 2, v.z);
        atomicAdd(dst + 3, v.w);
        if (c4 == 0) atomicAdd(&cnts[g], 1.0f);
    }
}
__global__ void k_final(const float* __restrict__ sums, const float* __restrict__ cnts,
                        const float* __restrict__ Wl, const float* __restrict__ bl,
                        float* __restrict__ out) {
    int idx = threadIdx.x;                     // 128 threads: 64 graphs x 2 outputs
    int g = idx >> 1, o = idx & 1;
    float inv = 1.0f / fmaxf(cnts[g], 1.0f);
    float acc = bl[o];
    for (int c = 0; c < HID; ++c) acc += sums[g * HID + c] * inv * Wl[c * 2 + o];
    out[idx] = acc;
}

extern "C" void kernel_launch(void* const* d_in, const int* in_sizes, int n_in,
                              void* d_out, int out_size, void* d_ws, size_t ws_size,
                              hipStream_t stream) {
    const float* x     = (const float*)d_in[0];
    const float* W1    = (const float*)d_in[1];
    const float* b1    = (const float*)d_in[2];
    const float* W2    = (const float*)d_in[3];
    const float* b2    = (const float*)d_in[4];
    const float* W3    = (const float*)d_in[5];
    const float* b3    = (const float*)d_in[6];
    const float* Wlin  = (const float*)d_in[7];
    const float* blin  = (const float*)d_in[8];
    const int*   ei    = (const int*)d_in[9];   // [2 x NE]: row then col
    const int*   batch = (const int*)d_in[10];
    float* out = (float*)d_out;

    const int* erow = ei;
    const int* ecol = ei + NE;

    // workspace carve-up (256B aligned)
    char* ws = (char*)d_ws;
    size_t off = 0;
    auto carve = [&](size_t bytes) -> char* {
        char* p = ws + off;
        off = (off + bytes + 255) & ~(size_t)255;
        return p;
    };
    float* dinv = (float*)carve((size_t)NN * 4);
    float* W1t  = (float*)carve((size_t)HID * CIN * 4);
    float* W2t  = (float*)carve((size_t)HID * HID * 4);
    float* W3t  = (float*)carve((size_t)HID * HID * 4);
    float* hw   = (float*)carve((size_t)NN * HID * 4);
    float* hA   = (float*)carve((size_t)NN * HID * 4);   // ping
    float* hB   = (float*)carve((size_t)NN * HID * 4);   // pong
    float* sums = (float*)carve((size_t)NG * HID * 4);
    float* cnts = (float*)carve((size_t)NG * 4);

    const int totWaves = (NN / 16) * 4;                 // 3125 row tiles x 4 col groups
    const int gemmBlocks = (totWaves + 7) / 8;
    const int edgeBlocks = 8192;                        // grid-stride, 4 edges/block/iter
    const int poolBlocks = 4096;

    // normalization coefficients
    k_deg_init <<<(NN + 255) / 256, 256, 0, stream>>>(dinv);
    k_deg_edges<<<(NE + 255) / 256, 256, 0, stream>>>(ecol, dinv);
    k_dinv     <<<(NN + 255) / 256, 256, 0, stream>>>(dinv);

    // weight transposes
    k_wt<<<(CIN * HID + 255) / 256, 256, 0, stream>>>(W1, W1t, CIN, HID);
    k_wt<<<(HID * HID + 255) / 256, 256, 0, stream>>>(W2, W2t, HID, HID);
    k_wt<<<(HID * HID + 255) / 256, 256, 0, stream>>>(W3, W3t, HID, HID);

    // layer 1: hw = x @ W1 ; hA = d^2*hw + b1 ; hA += edge scatter
    k_gemm_fused<CIN, 0><<<gemmBlocks, 256, 0, stream>>>(x, W1t, dinv, b1, hw, hA, totWaves);
    k_edge_agg<<<edgeBlocks, 256, 0, stream>>>(erow, ecol, dinv, hw, hA);

    // layer 2: hw = relu(hA) @ W2 ; hB = d^2*hw + b2 ; hB += edge scatter
    k_gemm_fused<HID, 1><<<gemmBlocks, 256, 0, stream>>>(hA, W2t, dinv, b2, hw, hB, totWaves);
    k_edge_agg<<<edgeBlocks, 256, 0, stream>>>(erow, ecol, dinv, hw, hB);

    // layer 3: hw = relu(hB) @ W3 ; hA = d^2*hw + b3 ; hA += edge scatter (no relu after)
    k_gemm_fused<HID, 1><<<gemmBlocks, 256, 0, stream>>>(hB, W3t, dinv, b3, hw, hA, totWaves);
    k_edge_agg<<<edgeBlocks, 256, 0, stream>>>(erow, ecol, dinv, hw, hA);

    // mean pool + linear head
    k_pool_zero<<<(NG * HID + 255) / 256, 256, 0, stream>>>(sums, cnts);
    k_pool_sum <<<poolBlocks, 256, 0, stream>>>(batch, hA, sums, cnts);
    k_final    <<<1, 128, 0, stream>>>(sums, cnts, Wlin, blin, out);
}